// FocalLoss_37263136260803
// MI455X (gfx1250) — compile-verified
//
#include <hip/hip_runtime.h>
#include <stdint.h>

// Problem constants from the reference: inputs [16,19,512,512] f32, targets [16,512,512] i32.
#define BLOCK      256
#define TILE_PX    256
#define CCH        19
#define HW_        (512*512)            // 262144 pixels per image (divisible by TILE_PX)
#define NPIX       (16*HW_)             // 4,194,304 pixels
#define NTILES     (NPIX / TILE_PX)     // 16384
#define NBLOCKS    1024
#define TPB        (NTILES / NBLOCKS)   // 16 tiles per block
#define CHUNKS     (CCH * TILE_PX / 4)  // 1216 x 16B chunks per tile
#define CPT        5                    // async issues per thread per tile (tail clamped)
#define NBUF       3                    // triple buffering: 2 tiles of prefetch in flight
#define IGNORE_LBL 255

// Issue one tile's worth of async global->LDS copies (ASYNCcnt-tracked, NT hint:
// 319 MB stream > 192 MB L2, zero reuse -> do not retain).
// LDS layout: channel-major, lds[c*TILE_PX + pixel]. Chunk k covers channel k>>6,
// floats [(k&63)*4, +4). Every thread issues exactly CPT instructions (clamped tail
// re-copies chunk 1215 with identical data -> benign, keeps s_wait immediates uniform).
__device__ __forceinline__ void async_tile(const float* __restrict__ inp,
                                           int tile, uint32_t lds_buf_off, int tid) {
  const int P0  = tile * TILE_PX;
  const int b   = P0 >> 18;             // image index (HW_ = 2^18)
  const int hw0 = P0 & (HW_ - 1);       // pixel offset within image
  const size_t gbase = ((size_t)(b * CCH) << 18) + (size_t)hw0;
#pragma unroll
  for (int m = 0; m < CPT; ++m) {
    int k = tid + m * BLOCK;
    k = (k < CHUNKS) ? k : (CHUNKS - 1);
    const int c = k >> 6;
    const int j = (k & 63) << 2;
    const uint64_t ga = (uint64_t)(uintptr_t)(inp + gbase + ((size_t)c << 18) + (size_t)j);
    const uint32_t la = lds_buf_off + (uint32_t)k * 16u;
    asm volatile("global_load_async_to_lds_b128 %0, %1, off th:TH_LOAD_NT"
                 :: "v"(la), "v"(ga) : "memory");
  }
}

__global__ __launch_bounds__(BLOCK) void focal_partial(const float* __restrict__ inp,
                                                       const int* __restrict__ tgt,
                                                       float* __restrict__ part) {
  __shared__ float lds[NBUF * CCH * TILE_PX];   // 3 x 19 KB staging buffers
  __shared__ float sF[BLOCK / 32], sC[BLOCK / 32];

  const int tid   = threadIdx.x;
  const int tile0 = blockIdx.x * TPB;
  const uint32_t lds0 = (uint32_t)(uintptr_t)&lds[0];   // low 32 bits = LDS byte offset
  const uint32_t BUFB = (uint32_t)(CCH * TILE_PX * 4);  // bytes per buffer

  // Prologue: prefetch tiles 0 and 1 into buffers 0 and 1.
  async_tile(inp, tile0 + 0, lds0 + 0u * BUFB, tid);
  async_tile(inp, tile0 + 1, lds0 + 1u * BUFB, tid);

  float accF = 0.f, accC = 0.f;
  int bi = 0;          // buffer holding tile i
  int bn = 2;          // buffer to receive tile i+2

  for (int i = 0; i < TPB; ++i) {
    if (i + 2 < TPB) {
      // Keep two tiles in flight: issue tile i+2, then wait until only those
      // 2*CPT issues remain (loads retire in order => tile i complete).
      async_tile(inp, tile0 + i + 2, lds0 + (uint32_t)bn * BUFB, tid);
      asm volatile("s_wait_asynccnt 0xa" ::: "memory");
    } else if (i + 1 < TPB) {
      asm volatile("s_wait_asynccnt 0x5" ::: "memory");
    } else {
      asm volatile("s_wait_asynccnt 0x0" ::: "memory");
    }
    __syncthreads();   // all waves' tile-i data visible in LDS

    const int gp = (tile0 + i) * TILE_PX + tid;
    const int t  = __builtin_nontemporal_load(tgt + gp);
    const bool valid = (t != IGNORE_LBL);

    // Bank-conflict-free: addr/4 = c*256 + tid -> bank = tid % 64, distinct per lane.
    const float* xb = &lds[bi * (CCH * TILE_PX)] + tid;
    float x[CCH];
#pragma unroll
    for (int c = 0; c < CCH; ++c) x[c] = xb[(size_t)c * TILE_PX];

    float mx = x[0];
#pragma unroll
    for (int c = 1; c < CCH; ++c) mx = fmaxf(mx, x[c]);

    float s = 0.f;
#pragma unroll
    for (int c = 0; c < CCH; ++c) s += __expf(x[c] - mx);
    const float lse = mx + __logf(s);

    const int tc = valid ? t : 0;
    float xt = x[0];
#pragma unroll
    for (int c = 1; c < CCH; ++c) xt = (c == tc) ? x[c] : xt;   // cndmask chain, no scratch

    const float ce    = lse - xt;
    const float pt    = __expf(-ce);
    const float focal = (1.f - pt) * ce;
    if (valid) { accF += focal; accC += 1.f; }

    __syncthreads();   // release buffer bi before it is refilled (iter i+1 issues into it)

    bi = (bi + 1 == NBUF) ? 0 : bi + 1;
    bn = (bn + 1 == NBUF) ? 0 : bn + 1;
  }

  // Deterministic block reduction: wave32 shuffle tree, then fixed-order LDS sum.
#pragma unroll
  for (int off = 16; off > 0; off >>= 1) {
    accF += __shfl_down(accF, off, 32);
    accC += __shfl_down(accC, off, 32);
  }
  if ((tid & 31) == 0) { sF[tid >> 5] = accF; sC[tid >> 5] = accC; }
  __syncthreads();
  if (tid == 0) {
    float F = 0.f, Cn = 0.f;
    for (int w = 0; w < BLOCK / 32; ++w) { F += sF[w]; Cn += sC[w]; }
    part[2 * blockIdx.x]     = F;
    part[2 * blockIdx.x + 1] = Cn;
  }
}

__global__ __launch_bounds__(BLOCK) void focal_final(const float* __restrict__ part,
                                                     float* __restrict__ out) {
  __shared__ float sF[BLOCK / 32], sC[BLOCK / 32];
  const int tid = threadIdx.x;
  float F = 0.f, Cn = 0.f;
  for (int b = tid; b < NBLOCKS; b += BLOCK) {   // fixed order per thread
    F  += part[2 * b];
    Cn += part[2 * b + 1];
  }
#pragma unroll
  for (int off = 16; off > 0; off >>= 1) {
    F  += __shfl_down(F, off, 32);
    Cn += __shfl_down(Cn, off, 32);
  }
  if ((tid & 31) == 0) { sF[tid >> 5] = F; sC[tid >> 5] = Cn; }
  __syncthreads();
  if (tid == 0) {
    float f = 0.f, c = 0.f;
    for (int w = 0; w < BLOCK / 32; ++w) { f += sF[w]; c += sC[w]; }
    out[0] = f / fmaxf(c, 1e-8f);
  }
}

extern "C" void kernel_launch(void* const* d_in, const int* in_sizes, int n_in,
                              void* d_out, int out_size, void* d_ws, size_t ws_size,
                              hipStream_t stream) {
  const float* inp = (const float*)d_in[0];   // [16,19,512,512] f32 logits
  const int*   tgt = (const int*)d_in[1];     // [16,512,512] i32 targets
  float* out  = (float*)d_out;                // scalar f32
  float* part = (float*)d_ws;                 // 2*NBLOCKS floats = 8 KB scratch

  focal_partial<<<NBLOCKS, BLOCK, 0, stream>>>(inp, tgt, part);
  focal_final<<<1, BLOCK, 0, stream>>>(part, out);

  (void)in_sizes; (void)n_in; (void)out_size; (void)ws_size;
}